// EdgePolicyModel_58652073394512
// MI455X (gfx1250) — compile-verified
//
#include <hip/hip_runtime.h>
#include <hip/hip_bf16.h>
#include <math.h>

// ---------------------------------------------------------------------------
// 2-layer GCN + edge readout for MI455X (gfx1250, wave32).
//   GEMMs (x@W1, h@W2): V_WMMA_F32_16X16X4_F32, 16x16 tiles, K=64.
//   Weight stage-in: GLOBAL_LOAD_ASYNC_TO_LDS_B128 (ASYNCcnt-tracked).
//   Aggregation: per-edge gather + f32 global atomics (L2-resident, 192MB L2).
//   Self-loops folded into the finalize pass: h = relu(acc + dinv^2*xw + b).
//   GEMM outputs are padded to a multiple of 128 rows -> unconditional stores
//   (no exec-mask predication in the epilogue).
// ---------------------------------------------------------------------------

typedef __attribute__((ext_vector_type(2))) float v2f;
typedef __attribute__((ext_vector_type(8))) float v8f;
typedef __attribute__((ext_vector_type(4))) int v4i;

#define CCH 64  // channels (F == C == 64)

#if defined(__has_builtin)
#if __has_builtin(__builtin_amdgcn_global_load_async_to_lds_b128) && \
    __has_builtin(__builtin_amdgcn_s_wait_asynccnt)
#define USE_ASYNC_LDS 1
#endif
#endif

// ---- init: acc = 0, deg = 1 (self loop), counter = 0 ----------------------
__global__ __launch_bounds__(256) void init_k(float* __restrict__ acc,
                                              float* __restrict__ deg,
                                              int* __restrict__ counter, int n) {
    int i = blockIdx.x * 256 + threadIdx.x;
    long total = (long)n * CCH;
    if (i < total) acc[i] = 0.0f;
    if (i < n) deg[i] = 1.0f;
    if (i == 0) counter[0] = 0;
}

// ---- deg[dst] += 1 per edge ----------------------------------------------
__global__ __launch_bounds__(256) void deg_count_k(const int* __restrict__ dst,
                                                   float* __restrict__ deg, int E) {
    int e = blockIdx.x * 256 + threadIdx.x;
    if (e < E) atomicAdd(&deg[dst[e]], 1.0f);
}

// ---- dinv = rsqrt(deg) ----------------------------------------------------
__global__ __launch_bounds__(256) void rsqrt_k(const float* __restrict__ deg,
                                               float* __restrict__ dinv, int n) {
    int i = blockIdx.x * 256 + threadIdx.x;
    if (i < n) {
        float d = deg[i];
        dinv[i] = (d > 0.0f) ? rsqrtf(d) : 0.0f;
    }
}

// ---- O[Np,64] = A[N,64] @ W[64,64] via WMMA f32 16x16x4 -------------------
// 256 threads = 8 waves; wave w owns rows [blk*128 + w*16, +16).
// Per wave: 4 column tiles x 16 k-steps = 64 v_wmma_f32_16x16x4_f32.
// O has >= gridDim.x*128 rows (padded), so stores need no predication.
__global__ __launch_bounds__(256) void gemm_n64_k(const float* __restrict__ A,
                                                  const float* __restrict__ W,
                                                  float* __restrict__ O, int n) {
    __shared__ __align__(16) float sW[CCH * CCH];
#ifdef USE_ASYNC_LDS
    // 1024 x 16B chunks; 256 lanes -> 4 async copies per lane, ASYNCcnt-tracked.
    for (int i = threadIdx.x; i < (CCH * CCH) / 4; i += 256) {
        __attribute__((address_space(1))) v4i* g =
            (__attribute__((address_space(1))) v4i*)(W + 4 * i);
        __attribute__((address_space(3))) v4i* l =
            (__attribute__((address_space(3))) v4i*)(&sW[4 * i]);
        __builtin_amdgcn_global_load_async_to_lds_b128(g, l, 0, 0);
    }
    __builtin_amdgcn_s_wait_asynccnt(0);
#else
    for (int i = threadIdx.x; i < CCH * CCH; i += 256) sW[i] = W[i];
#endif
    __syncthreads();

    const int wave = threadIdx.x >> 5;
    const int lane = threadIdx.x & 31;
    const int lhalf = lane >> 4;   // 0: lanes 0-15, 1: lanes 16-31
    const int lmod = lane & 15;
    const int rowBase = blockIdx.x * 128 + wave * 16;

    // clamp row for loads only (A may have exactly n rows); EXEC stays all-1s.
    int r = rowBase + lmod;
    int rr = (r < n) ? r : (n - 1);
    const float* arow = A + (size_t)rr * CCH;

    // Preload all 16 A fragments (k-step s covers K = 4s..4s+3):
    // lane layout (ISA 32-bit A 16x4): VGPR0/1 hold K = 2*lhalf + {0,1}.
    v2f afrag[16];
#pragma unroll
    for (int s = 0; s < 16; ++s) {
        int kidx = 4 * s + 2 * lhalf;
        afrag[s].x = arow[kidx];
        afrag[s].y = arow[kidx + 1];
    }

#pragma unroll
    for (int nt = 0; nt < 4; ++nt) {
        v8f c = {0.f, 0.f, 0.f, 0.f, 0.f, 0.f, 0.f, 0.f};
#pragma unroll
        for (int s = 0; s < 16; ++s) {
            int kidx = 4 * s + 2 * lhalf;
            v2f b;
            b.x = sW[kidx * CCH + nt * 16 + lmod];
            b.y = sW[(kidx + 1) * CCH + nt * 16 + lmod];
            c = __builtin_amdgcn_wmma_f32_16x16x4_f32(
                /*neg_a=*/false, afrag[s], /*neg_b=*/false, b,
                /*c_mod=*/(short)0, c, /*reuse_a=*/false, /*reuse_b=*/false);
        }
        // D layout: VGPR i -> row rowBase + 8*lhalf + i, col nt*16 + lmod.
        // O is row-padded; store unconditionally (no exec predication).
#pragma unroll
        for (int i = 0; i < 8; ++i) {
            int row = rowBase + 8 * lhalf + i;
            O[(size_t)row * CCH + nt * 16 + lmod] = c[i];
        }
    }
}

// ---- per-edge scatter: acc[dst] += dinv[src]*dinv[dst] * xw[src] ----------
// One wave (32 lanes) per edge; each lane moves 2 channels (float2).
__global__ __launch_bounds__(256) void edge_scatter_k(const int* __restrict__ src,
                                                      const int* __restrict__ dst,
                                                      const float* __restrict__ dinv,
                                                      const float* __restrict__ xw,
                                                      float* __restrict__ acc, int E) {
    int e = blockIdx.x * 8 + (threadIdx.x >> 5);
    if (e >= E) return;
    int lane = threadIdx.x & 31;
    int s = src[e], d = dst[e];
    float nrm = dinv[s] * dinv[d];
    const float2* xs = (const float2*)(xw + (size_t)s * CCH);
    float2 v = xs[lane];
    float* ad = acc + (size_t)d * CCH + 2 * lane;
    atomicAdd(ad, v.x * nrm);
    atomicAdd(ad + 1, v.y * nrm);
}

// ---- finalize: h = relu(acc + dinv^2 * xw + b); acc reset to 0 ------------
__global__ __launch_bounds__(256) void finalize_k(const float* __restrict__ xw,
                                                  float* __restrict__ acc,
                                                  const float* __restrict__ dinv,
                                                  const float* __restrict__ b,
                                                  float* __restrict__ h, int n) {
    long i = (long)blockIdx.x * 256 + threadIdx.x;
    if (i >= (long)n * CCH) return;
    int v = (int)(i >> 6), cch = (int)(i & 63);
    float di = dinv[v];
    float val = acc[i] + di * di * xw[i] + b[cch];
    h[i] = val > 0.0f ? val : 0.0f;
    acc[i] = 0.0f;  // ready for next layer
}

// ---- collect edges touching `node` (set is deterministic; order fixed later)
__global__ __launch_bounds__(256) void collect_k(const int* __restrict__ s0,
                                                 const int* __restrict__ d0, int E,
                                                 const int* __restrict__ nodep,
                                                 int* __restrict__ idxbuf,
                                                 int* __restrict__ counter) {
    int e = blockIdx.x * 256 + threadIdx.x;
    if (e >= E) return;
    int node = nodep[0];
    if (s0[e] == node || d0[e] == node) {
        int p = atomicAdd(counter, 1);
        if (p < 64) idxbuf[p] = e;
    }
}

// ---- readout: order by edge id, score = softmax(h[targets] @ Wr + br) -----
__global__ __launch_bounds__(64) void readout_k(const int* __restrict__ idxbuf,
                                                const int* __restrict__ s0,
                                                const int* __restrict__ d0,
                                                const int* __restrict__ nodep,
                                                const float* __restrict__ h,
                                                const float* __restrict__ Wr,
                                                const float* __restrict__ br,
                                                float* __restrict__ out) {
    __shared__ int ses[64];
    __shared__ float sval[64];
    __shared__ int stgt[64];
    int t = threadIdx.x;
    int node = nodep[0];
    int e = idxbuf[t];
    ses[t] = e;
    __syncthreads();

    // rank = position in ascending edge order (matches jnp.nonzero ordering)
    int rank = 0;
    for (int j = 0; j < 64; ++j) rank += (ses[j] < e) ? 1 : 0;

    int s = s0[e], d = d0[e];
    int target = (s == node) ? d : s;  // self-loop (s==d==node) -> node

    float acc = br[0];
    const float* hrow = h + (size_t)target * CCH;
    for (int c = 0; c < CCH; ++c) acc += hrow[c] * Wr[c];
    sval[rank] = acc;
    stgt[rank] = target;
    __syncthreads();

    // softmax over the 64 scores (every thread scans all; deterministic order)
    float m = -INFINITY;
    for (int j = 0; j < 64; ++j) m = fmaxf(m, sval[j]);
    float sum = 0.0f;
    for (int j = 0; j < 64; ++j) sum += expf(sval[j] - m);
    out[t] = expf(sval[t] - m) / sum;
    out[64 + t] = (float)stgt[t];
}

// ---------------------------------------------------------------------------
extern "C" void kernel_launch(void* const* d_in, const int* in_sizes, int n_in,
                              void* d_out, int out_size, void* d_ws, size_t ws_size,
                              hipStream_t stream) {
    const float* x  = (const float*)d_in[0];
    const int*   ei = (const int*)d_in[1];   // [2, E] flattened
    const int*   nodep = (const int*)d_in[2];
    const float* W1 = (const float*)d_in[3];
    const float* b1 = (const float*)d_in[4];
    const float* W2 = (const float*)d_in[5];
    const float* b2 = (const float*)d_in[6];
    const float* Wr = (const float*)d_in[7];
    const float* br = (const float*)d_in[8];
    float* out = (float*)d_out;

    const int N = in_sizes[0] / CCH;
    const int E = in_sizes[1] / 2;
    const int Np = (N + 127) & ~127;         // GEMM row padding (128-row tiles)
    const int* srcs = ei;
    const int* dsts = ei + E;

    // workspace carve (all fp32 word aligned; acc/xw/h padded to Np rows)
    float* acc  = (float*)d_ws;
    float* xw   = acc + (size_t)Np * CCH;
    float* h    = xw + (size_t)Np * CCH;
    float* deg  = h + (size_t)Np * CCH;
    float* dinv = deg + N;
    int* counter = (int*)(dinv + N);
    int* idxbuf  = counter + 1;

    const long NC = (long)N * CCH;
    dim3 b256(256);
    dim3 gNC((unsigned)((NC + 255) / 256));
    dim3 gN((unsigned)((N + 255) / 256));
    dim3 gE((unsigned)((E + 255) / 256));
    dim3 gE8((unsigned)((E + 7) / 8));       // 8 edges/block (1 wave per edge)
    dim3 gGemm((unsigned)(Np / 128));        // 128 rows/block (8 waves x 16)

    // degree + normalization
    init_k<<<gNC, b256, 0, stream>>>(acc, deg, counter, N);
    deg_count_k<<<gE, b256, 0, stream>>>(dsts, deg, E);
    rsqrt_k<<<gN, b256, 0, stream>>>(deg, dinv, N);

    // layer 1: xw = x@W1 ; acc = scatter ; h = relu(acc + dinv^2*xw + b1)
    gemm_n64_k<<<gGemm, b256, 0, stream>>>(x, W1, xw, N);
    edge_scatter_k<<<gE8, b256, 0, stream>>>(srcs, dsts, dinv, xw, acc, E);
    finalize_k<<<gNC, b256, 0, stream>>>(xw, acc, dinv, b1, h, N);

    // layer 2 (reuses xw and acc buffers; finalize_k zeroed acc)
    gemm_n64_k<<<gGemm, b256, 0, stream>>>(h, W2, xw, N);
    edge_scatter_k<<<gE8, b256, 0, stream>>>(srcs, dsts, dinv, xw, acc, E);
    finalize_k<<<gNC, b256, 0, stream>>>(xw, acc, dinv, b2, h, N);

    // readout over the K=64 edges incident to `node`
    collect_k<<<gE, b256, 0, stream>>>(srcs, dsts, E, nodep, idxbuf, counter);
    readout_k<<<dim3(1), dim3(64), 0, stream>>>(idxbuf, srcs, dsts, nodep, h, Wr, br, out);
}